// InverseWaveletBlock1D_64415919505833
// MI455X (gfx1250) — compile-verified
//
#include <hip/hip_runtime.h>
#include <hip/hip_bf16.h>

// ---------------------------------------------------------------------------
// Inverse 1D wavelet synthesis, 4 levels, 64 channels. Memory-bound
// (~180 MB total traffic @ 23.3 TB/s => ~8us floor). 4 ping-pong passes,
// polyphase conv as X(16x20) * W_banded(20x16) via chained
// V_WMMA_F32_16X16X4_F32 (native f32 WMMA, wave32).
//
// Round-1 fixes: boundary handling is now a *uniform* per-subtile branch
// (interior tiles stage with one unconditional coalesced load), and the
// banded filter fragments are branchless cndmask selects hoisted out of the
// K loop and reused across 4 position sub-tiles per wave.
// ---------------------------------------------------------------------------

typedef float v2f __attribute__((ext_vector_type(2)));
typedef float v8f __attribute__((ext_vector_type(8)));

// Branchless band coefficient: W[r][n] = c0,c1,c2 for r-n = 0,1,2, else 0.
// Inputs are registers -> compiler emits v_cmp + v_cndmask, no branches.
__device__ __forceinline__ float bandsel(int d, float c0, float c1, float c2) {
    float r = 0.0f;
    r = (d == 0) ? c0 : r;
    r = (d == 1) ? c1 : r;
    r = (d == 2) ? c2 : r;
    return r;
}

// Slow-path staged element for boundary tiles only (2 sub-tiles per level).
__device__ __forceinline__ float load_xp_slow(const float* __restrict__ row, int gi, int L) {
    if (gi >= 1 && gi <= L)  return row[gi - 1];
    if (gi == 0)             return 2.0f * row[0]     - row[1];
    if (gi == L + 1)         return 2.0f * row[L - 1] - row[L - 2];
    return 0.0f;
}

// One synthesis level. grid = L/64 blocks of 128 threads (4 waves).
// Wave w owns channels [16w,16w+16); each wave processes 4 sub-tiles of 16
// output positions: t0 = 64*blockIdx.x + 16*it.
__global__ __launch_bounds__(128) void synth_level_kernel(
        const float* __restrict__ x, const float* __restrict__ d,
        float* __restrict__ out,
        const float* __restrict__ h, const float* __restrict__ g,
        int L) {
    __shared__ float smx[4][16][20];
    __shared__ float smd[4][16][20];

    const int w    = threadIdx.x >> 5;
    const int lane = threadIdx.x & 31;
    const int c0   = w << 4;
    const int n    = lane & 15;          // N (position) / A-matrix M (channel) lane index
    const int hi   = lane >> 4;          // half-wave: K offset +2 for A and B

    // ---- taps (uniform, L2-cached) ----
    const float h0 = h[0], h1 = h[1], h2 = h[2], h3 = h[3], h4 = h[4], h5 = h[5];
    const float g0 = g[0], g1 = g[1], g2 = g[2], g3 = g[3], g4 = g[4], g5 = g[5];

    // ---- B fragments: depend only on lane, hoisted out of all loops ----
    // B 4x16 layout: VGPR r holds row (chunk + r) low lanes / (+2) high lanes, col n.
    v2f bhe[5], bho[5], bge[5], bgo[5];
#pragma unroll
    for (int c = 0; c < 5; ++c) {
        const int r0 = c * 4 + (hi << 1);
        const int d0r = r0 - n, d1r = r0 + 1 - n;
        bhe[c] = v2f{ bandsel(d0r, h4, h2, h0), bandsel(d1r, h4, h2, h0) };
        bho[c] = v2f{ bandsel(d0r, h5, h3, h1), bandsel(d1r, h5, h3, h1) };
        bge[c] = v2f{ bandsel(d0r, g4, g2, g0), bandsel(d1r, g4, g2, g0) };
        bgo[c] = v2f{ bandsel(d0r, g5, g3, g1), bandsel(d1r, g5, g3, g1) };
    }

    const size_t outL = 2 * (size_t)L;

    for (int it = 0; it < 4; ++it) {
        const int t0 = (blockIdx.x << 6) + (it << 4);

        // ---- stage xp/dp windows [t0 .. t0+19] for 16 channel rows ----
        // Interior test is uniform per block (s_cbranch, no divergence).
        if (t0 >= 1 && t0 + 19 <= L) {
            // fast path: window maps to row[t0-1 .. t0+18], unconditional loads
            for (int idx = lane; idx < 16 * 20; idx += 32) {
                const int r = idx / 20;
                const int i = idx % 20;
                const size_t rowBase = (size_t)(c0 + r) * (size_t)L + (size_t)(t0 - 1);
                smx[w][r][i] = x[rowBase + i];
                smd[w][r][i] = d[rowBase + i];
            }
        } else {
            // boundary path (first/last sub-tile of the level only)
            for (int idx = lane; idx < 16 * 20; idx += 32) {
                const int r  = idx / 20;
                const int i  = idx % 20;
                const int gi = t0 + i;
                const size_t rowBase = (size_t)(c0 + r) * (size_t)L;
                smx[w][r][i] = load_xp_slow(x + rowBase, gi, L);
                smd[w][r][i] = load_xp_slow(d + rowBase, gi, L);
            }
        }
        __syncthreads();

        // ---- out(16x16) = X(16x20)*W(20x16), even+odd phase, 2 branches ----
        v8f ce = {};
        v8f co = {};
#pragma unroll
        for (int c = 0; c < 5; ++c) {
            const int k0 = c * 4 + (hi << 1);   // A 16x4: VGPR0/1 = K k0,k0+1
            const v2f ax = v2f{ smx[w][n][k0], smx[w][n][k0 + 1] };
            const v2f ad = v2f{ smd[w][n][k0], smd[w][n][k0 + 1] };

            ce = __builtin_amdgcn_wmma_f32_16x16x4_f32(false, ax, false, bhe[c], (short)0, ce, false, false);
            ce = __builtin_amdgcn_wmma_f32_16x16x4_f32(false, ad, false, bge[c], (short)0, ce, false, false);
            co = __builtin_amdgcn_wmma_f32_16x16x4_f32(false, ax, false, bho[c], (short)0, co, false, false);
            co = __builtin_amdgcn_wmma_f32_16x16x4_f32(false, ad, false, bgo[c], (short)0, co, false, false);
        }

        // ---- store interleaved even/odd as float2 (contiguous per half-wave) ----
        const int pos = t0 + n;
#pragma unroll
        for (int r = 0; r < 8; ++r) {
            const int ch = c0 + r + (hi << 3);  // C/D: VGPR r -> M=r (+8 high half)
            float2 v;
            v.x = ce[r];
            v.y = co[r];
            *reinterpret_cast<float2*>(&out[(size_t)ch * outL + 2 * (size_t)pos]) = v;
        }
        __syncthreads();
    }
}

extern "C" void kernel_launch(void* const* d_in, const int* in_sizes, int n_in,
                              void* d_out, int out_size, void* d_ws, size_t ws_size,
                              hipStream_t stream) {
    (void)in_sizes; (void)n_in; (void)out_size; (void)ws_size;

    const float* sig = (const float*)d_in[0];   // (64,1,16384)
    const float* d0  = (const float*)d_in[1];   // (64,1,131072)
    const float* d1  = (const float*)d_in[2];   // (64,1,65536)
    const float* d2  = (const float*)d_in[3];   // (64,1,32768)
    const float* d3  = (const float*)d_in[4];   // (64,1,16384)
    const float* h   = (const float*)d_in[5];   // (6,)
    const float* g   = (const float*)d_in[6];   // (6,)
    float* out = (float*)d_out;                 // (64,1,262144)

    // ping-pong scratch: A holds the 65536/ch level, B holds 32768/ch then 131072/ch
    float* A = (float*)d_ws;                        // 64*65536 floats (16 MiB)
    float* B = A + (size_t)64 * 65536;              // 64*131072 floats (32 MiB)

    dim3 blk(128);
    synth_level_kernel<<<16384  / 64, blk, 0, stream>>>(sig, d3, B,   h, g, 16384);
    synth_level_kernel<<<32768  / 64, blk, 0, stream>>>(B,   d2, A,   h, g, 32768);
    synth_level_kernel<<<65536  / 64, blk, 0, stream>>>(A,   d1, B,   h, g, 65536);
    synth_level_kernel<<<131072 / 64, blk, 0, stream>>>(B,   d0, out, h, g, 131072);
}